// GNNModel_21698174780131
// MI455X (gfx1250) — compile-verified
//
#include <hip/hip_runtime.h>
#include <math.h>

typedef __attribute__((ext_vector_type(2))) float v2f;
typedef __attribute__((ext_vector_type(8))) float v8f;

#define D_FEAT       128
#define N_PER_GRAPH  100000
#define NUM_GRAPHS   8
#define TOTAL_NODES  (N_PER_GRAPH * NUM_GRAPHS)   // 800000
#define NUM_EDGES    25600000

// ---------------------------------------------------------------------------
// Kernel 1: init integer degree (self-loop => 1) and zero the graph sums.
// ---------------------------------------------------------------------------
__global__ void k_init(unsigned* __restrict__ cnt, float* __restrict__ gsum) {
    int v = blockIdx.x * blockDim.x + threadIdx.x;
    if (v < TOTAL_NODES) cnt[v] = 1u;
    if (v < NUM_GRAPHS)  gsum[v] = 0.0f;
}

// ---------------------------------------------------------------------------
// Kernel 2: h = x @ W + b via V_WMMA_F32_16X16X4_F32.
// One wave handles a 16-row tile of x; K=128 -> 32 WMMA steps of K=4.
// A layout (32-bit A 16x4): lane l -> M = l&15, VGPR0 = K(2*(l>>4)),
//                           VGPR1 = K(2*(l>>4)+1).
// B layout mirrors it with N = l&15; only column N==0 carries W.
// D layout: lane 0 holds D[M=r][0] in VGPR r (r=0..7), lane 16 holds M=8+r.
// ---------------------------------------------------------------------------
__global__ void __launch_bounds__(256) k_gemv_wmma(const float* __restrict__ x,
                                                   const float* __restrict__ W,
                                                   const float* __restrict__ bias,
                                                   float* __restrict__ h) {
    const int lane  = threadIdx.x & 31;
    const int wave  = threadIdx.x >> 5;
    const int tile  = blockIdx.x * 8 + wave;          // 16 rows per tile
    const int m     = lane & 15;                      // row within tile (A) / col (B)
    const int kh    = (lane >> 4) << 1;               // 0 or 2: K sub-pair
    const long row  = (long)tile * 16 + m;

    const float* __restrict__ xr = x + row * D_FEAT + kh;
    const float* __restrict__ wr = W + kh;

    v8f c = {};
#pragma unroll
    for (int k0 = 0; k0 < D_FEAT; k0 += 4) {
        v2f a, bv;
        a.x = xr[k0];
        a.y = xr[k0 + 1];
        float w0 = wr[k0];
        float w1 = wr[k0 + 1];
        bv.x = (m == 0) ? w0 : 0.0f;                  // B[:,0] = W, rest 0
        bv.y = (m == 0) ? w1 : 0.0f;
        c = __builtin_amdgcn_wmma_f32_16x16x4_f32(false, a, false, bv,
                                                  (short)0, c, false, false);
    }

    const float bb = bias[0];
    if (m == 0) {                                     // lanes 0 and 16 own column N=0
        float* hp = h + (long)tile * 16 + ((lane >> 4) ? 8 : 0);
#pragma unroll
        for (int r = 0; r < 8; ++r) hp[r] = c[r] + bb;
    }
}

// ---------------------------------------------------------------------------
// Kernel 3: degree accumulation over edges (integer atomics, exact).
// ---------------------------------------------------------------------------
__global__ void k_deg(const int* __restrict__ ei, unsigned* __restrict__ cnt) {
    int e   = blockIdx.x * blockDim.x + threadIdx.x;
    int dst = ei[NUM_EDGES + e];
    atomicAdd(&cnt[dst], 1u);
}

// ---------------------------------------------------------------------------
// Kernel 4: dis = rsqrt(deg); seed acc with self-loop term dis^2 * h.
// ---------------------------------------------------------------------------
__global__ void k_norm(const unsigned* __restrict__ cnt,
                       const float* __restrict__ h,
                       float* __restrict__ dis,
                       float* __restrict__ acc) {
    int v = blockIdx.x * blockDim.x + threadIdx.x;
    float d = rsqrtf((float)cnt[v]);                  // deg >= 1 always
    dis[v] = d;
    acc[v] = d * d * h[v];
}

// ---------------------------------------------------------------------------
// Kernel 5: scatter messages: acc[dst] += dis[src]*dis[dst]*h[src].
// ---------------------------------------------------------------------------
__global__ void k_msg(const int* __restrict__ ei,
                      const float* __restrict__ h,
                      const float* __restrict__ dis,
                      float* __restrict__ acc) {
    int e = blockIdx.x * blockDim.x + threadIdx.x;
    int s = ei[e];
    int d = ei[NUM_EDGES + e];
    float val = dis[s] * dis[d] * h[s];
    __hip_atomic_fetch_add(&acc[d], val, __ATOMIC_RELAXED, __HIP_MEMORY_SCOPE_AGENT);
}

// ---------------------------------------------------------------------------
// Kernel 6: ReLU * mask, pool into per-graph sums (LDS f32 atomics first,
// then <= 8 global atomics per block).
// ---------------------------------------------------------------------------
__global__ void __launch_bounds__(256) k_pool(const float* __restrict__ acc,
                                              const float* __restrict__ dec,
                                              float* __restrict__ gsum) {
    __shared__ float s[NUM_GRAPHS];
    int t = threadIdx.x;
    if (t < NUM_GRAPHS) s[t] = 0.0f;
    __syncthreads();

    int   v   = blockIdx.x * 256 + t;                 // 3125 * 256 == TOTAL_NODES
    float val = acc[v];
    val = val > 0.0f ? val : 0.0f;                    // ReLU
    val *= dec[v % N_PER_GRAPH];                      // tiled decision mask
    int g = v / N_PER_GRAPH;
    __hip_atomic_fetch_add(&s[g], val, __ATOMIC_RELAXED, __HIP_MEMORY_SCOPE_WORKGROUP);
    __syncthreads();

    if (t < NUM_GRAPHS)
        __hip_atomic_fetch_add(&gsum[t], s[t], __ATOMIC_RELAXED, __HIP_MEMORY_SCOPE_AGENT);
}

// ---------------------------------------------------------------------------
// Kernel 7: mean pool + log_softmax over a single class (-> exactly 0).
// ---------------------------------------------------------------------------
__global__ void k_final(const float* __restrict__ gsum, float* __restrict__ out) {
    int g = threadIdx.x;
    if (g < NUM_GRAPHS) {
        float pooled = gsum[g] * (1.0f / (float)N_PER_GRAPH);
        float mx  = pooled;                           // row max (1 element)
        float lse = mx + logf(expf(pooled - mx));     // logsumexp over axis=1
        out[g] = pooled - lse;
    }
}

// ---------------------------------------------------------------------------
extern "C" void kernel_launch(void* const* d_in, const int* in_sizes, int n_in,
                              void* d_out, int out_size, void* d_ws, size_t ws_size,
                              hipStream_t stream) {
    (void)in_sizes; (void)n_in; (void)out_size; (void)ws_size;

    const float* x   = (const float*)d_in[0];   // [800000, 128]
    const float* W   = (const float*)d_in[1];   // [128, 1]
    const float* b   = (const float*)d_in[2];   // [1]
    const float* dec = (const float*)d_in[3];   // [100000]
    const int*   ei  = (const int*)d_in[4];     // [2, 25600000]
    // d_in[5] = batch (derivable from node index), d_in[6] = num_graphs (8)

    float* out = (float*)d_out;                 // [8, 1]

    char*     ws   = (char*)d_ws;
    float*    h    = (float*)(ws);                                         // 3.2 MB
    unsigned* cnt  = (unsigned*)(ws + sizeof(float) * (size_t)TOTAL_NODES);      // 3.2 MB
    float*    dis  = (float*)(ws + sizeof(float) * (size_t)TOTAL_NODES * 2);     // 3.2 MB
    float*    acc  = (float*)(ws + sizeof(float) * (size_t)TOTAL_NODES * 3);     // 3.2 MB
    float*    gsum = (float*)(ws + sizeof(float) * (size_t)TOTAL_NODES * 4);     // 32 B

    k_init     <<<TOTAL_NODES / 256,        256, 0, stream>>>(cnt, gsum);
    k_gemv_wmma<<<TOTAL_NODES / (16 * 8),   256, 0, stream>>>(x, W, b, h);   // 6250 blocks
    k_deg      <<<NUM_EDGES / 256,          256, 0, stream>>>(ei, cnt);      // 100000 blocks
    k_norm     <<<TOTAL_NODES / 256,        256, 0, stream>>>(cnt, h, dis, acc);
    k_msg      <<<NUM_EDGES / 256,          256, 0, stream>>>(ei, h, dis, acc);
    k_pool     <<<TOTAL_NODES / 256,        256, 0, stream>>>(acc, dec, gsum);
    k_final    <<<1, 32, 0, stream>>>(gsum, out);
}